// Density_Softmax_70033736729140
// MI455X (gfx1250) — compile-verified
//
#include <hip/hip_runtime.h>
#include <hip/hip_bf16.h>

#define BB 128
#define CC 4096
#define DD 256
#define KK 256
#define EPSV 1e-8f
#define BIGV 1000.0f
#define THRESHV 0.2f

typedef __attribute__((ext_vector_type(2))) float v2f;
typedef __attribute__((ext_vector_type(8))) float v8f;

// ---------------- wnorm[c] = sum_d w[c,d]^2 ----------------
__global__ __launch_bounds__(256) void prep_norms(const float* __restrict__ weight,
                                                  float* __restrict__ wnorm) {
    int c = blockIdx.x * 256 + threadIdx.x;  // 4096 threads
    float s = 0.f;
    const float* row = weight + (size_t)c * DD;
    for (int d = 0; d < DD; ++d) { float w = row[d]; s += w * w; }
    wnorm[c] = s;
}

// ---------------- sw[b,:] = weight[labels[b],:], swnorm[b] ----------------
__global__ __launch_bounds__(256) void prep_sw(const float* __restrict__ weight,
                                               const int* __restrict__ labels,
                                               const float* __restrict__ wnorm,
                                               float* __restrict__ sw,
                                               float* __restrict__ swnorm) {
    int b = blockIdx.x, d = threadIdx.x;
    int lab = labels[b];
    sw[b * DD + d] = weight[(size_t)lab * DD + d];
    if (d == 0) swnorm[b] = wnorm[lab];
}

// ---------------- dis[b,c] via V_WMMA_F32_16X16X4_F32 ----------------
// One wave32 per 16x16 output tile. Tiles: (128/16) x (4096/16) = 8 x 256.
__global__ __launch_bounds__(32) void gemm_dis(const float* __restrict__ weight,
                                               const float* __restrict__ sw,
                                               const float* __restrict__ wnorm,
                                               const float* __restrict__ swnorm,
                                               float* __restrict__ dis) {
    const int tile = blockIdx.x;
    const int tn = tile & 255;       // column tile (C dim)
    const int tm = tile >> 8;        // row tile (B dim)
    const int lane = threadIdx.x;
    const int half = lane >> 4;      // 0: K=+0,+1 ; 1: K=+2,+3 (32-bit A/B layout)
    const int l = lane & 15;
    const int m0 = tm * 16, n0 = tn * 16;

    const float* arow = sw + (size_t)(m0 + l) * DD;      // A: M = l
    const float* brow = weight + (size_t)(n0 + l) * DD;  // B(K,N) = W[n0+l, k]

    v8f acc = {};
    #pragma unroll 4
    for (int k0 = 0; k0 < DD; k0 += 4) {
        const int ka = k0 + half * 2;
        v2f a = *(const v2f*)(arow + ka);
        v2f bfrag = *(const v2f*)(brow + ka);
        // 8 args: (neg_a, A, neg_b, B, c_mod, C, reuse_a, reuse_b)
        acc = __builtin_amdgcn_wmma_f32_16x16x4_f32(
            false, a, false, bfrag, (short)0, acc, false, false);
    }
    // C/D layout: VGPR r -> M = r + 8*half, N = l
    const int col = n0 + l;
    const float wn = wnorm[col];
    #pragma unroll
    for (int r = 0; r < 8; ++r) {
        const int row = m0 + r + 8 * half;
        dis[(size_t)row * CC + col] = swnorm[row] + wn - 2.0f * acc[r];
    }
}

// ---------------- bitonic full sort of each dis row; keep top-256 indices -----
// Descending by value, ties -> lower original index (matches jax.lax.top_k).
__global__ __launch_bounds__(512) void sort_topk(const float* __restrict__ dis,
                                                 int* __restrict__ idxbuf) {
    __shared__ float sv[CC];
    __shared__ int   si[CC];
    const int b = blockIdx.x, tid = threadIdx.x;
    for (int i = tid; i < CC; i += 512) { sv[i] = dis[(size_t)b * CC + i]; si[i] = i; }
    __syncthreads();
    for (int kk = 2; kk <= CC; kk <<= 1) {
        for (int j = kk >> 1; j > 0; j >>= 1) {
            for (int t = tid; t < CC / 2; t += 512) {
                const int i = ((t & ~(j - 1)) << 1) | (t & (j - 1));
                const int p = i | j;
                const bool dir = ((i & kk) == 0);   // true => descending segment
                float va = sv[i], vb = sv[p];
                int   ia = si[i], ib = si[p];
                const bool aFirst = (va > vb) || (va == vb && ia < ib);
                if (dir ? !aFirst : aFirst) {
                    sv[i] = vb; sv[p] = va; si[i] = ib; si[p] = ia;
                }
            }
            __syncthreads();
        }
    }
    for (int k = tid; k < KK; k += 512) idxbuf[b * KK + k] = si[k];
}

// ---------------- fused per-(b,d) statistics + tdd ----------------
// Native v_exp_f32/v_log_f32 via __expf/__logf: the 134M-exp tdd loop is the
// runtime dominator; weight (4MB) stays L2-resident so this is purely
// transcendental-throughput bound.
__global__ __launch_bounds__(256) void stats_kernel(const float* __restrict__ weight,
                                                    const float* __restrict__ sw,
                                                    const float* __restrict__ mu,
                                                    const float* __restrict__ var,
                                                    const int* __restrict__ labels,
                                                    const unsigned char* __restrict__ nontrivial,
                                                    const int* __restrict__ idxbuf,
                                                    float* __restrict__ L1v,
                                                    float* __restrict__ dfv,
                                                    float* __restrict__ mval,
                                                    float* __restrict__ totv,
                                                    float* __restrict__ gval) {
    __shared__ int sidx[KK];
    const int b = blockIdx.x, d = threadIdx.x;
    sidx[d] = idxbuf[b * KK + d];
    __syncthreads();

    const int lab = labels[b];
    const float mub   = mu[b * DD + d];
    const float inv2v = 0.5f / var[b * DD + d];
    const float swd   = sw[b * DD + d];
    const float dsw   = swd - mub;
    const float density = __expf(-dsw * dsw * inv2v);

    // pass 1: total + argmin (first-min tie break)
    float total = 0.f, tmin = 3.402823e38f;
    int amin = 0;
    #pragma unroll 4
    for (int k = 0; k < KK; ++k) {
        const float w  = weight[(size_t)sidx[k] * DD + d];
        const float df = w - mub;
        const float td = __expf(-df * df * inv2v);
        total += td;
        if (td < tmin) { tmin = td; amin = k; }
    }
    total = fmaxf(total, EPSV);

    // pass 2: masked min/max and gathered flag (recompute td)
    float minv = 3.402823e38f, maxv = -3.402823e38f, g = 0.f;
    const unsigned char* ntrow = nontrivial + (size_t)b * KK * DD + d;
    #pragma unroll 4
    for (int k = 0; k < KK; ++k) {
        const float w  = weight[(size_t)sidx[k] * DD + d];
        const float df = w - mub;
        const float td = __expf(-df * df * inv2v);
        const bool nt = (ntrow[(size_t)k * DD] != 0) && (k != amin);
        minv = fminf(minv, td + (nt ? 0.f : BIGV));
        maxv = fmaxf(maxv, td - (nt ? BIGV : 0.f));
        if (k == lab) g = nt ? 1.f : 0.f;
    }

    // tdd: sum over all C rows (hot loop: 6 VALU + 1 coalesced 4B L2 load / iter)
    float tdd = 0.f;
    #pragma unroll 8
    for (int c = 0; c < CC; ++c) {
        const float w  = weight[(size_t)c * DD + d];
        const float df = w - mub;
        tdd += __expf(-df * df * inv2v);
    }
    tdd = fmaxf(tdd, EPSV);

    const float ld = __logf(density);
    const float l1 = __logf(total) - ld;   // -log(density/total)
    const float l2 = ld - __logf(tdd);     //  log(density/tdd)
    L1v[b * DD + d]  = l1;
    dfv[b * DD + d]  = l2 - l1;
    mval[b * DD + d] = minv - maxv;
    totv[b * DD + d] = total;
    gval[b * DD + d] = g;
}

// ---------------- per-d reduction: sum_{b2} (B*L1 + diff*N[b2,d]) ----------------
__global__ __launch_bounds__(128) void reduce_d(const float* __restrict__ L1v,
                                                const float* __restrict__ dfv,
                                                const float* __restrict__ mval,
                                                const float* __restrict__ totv,
                                                const float* __restrict__ gval,
                                                float* __restrict__ partial) {
    __shared__ float ms[BB], gs[BB], red[BB];
    const int d = blockIdx.x, tid = threadIdx.x;   // tid = b index
    ms[tid] = mval[(size_t)tid * DD + d];
    gs[tid] = gval[(size_t)tid * DD + d];
    __syncthreads();
    const float t = THRESHV * totv[(size_t)tid * DD + d];
    int cnt = 0;
    #pragma unroll 4
    for (int b1 = 0; b1 < BB; ++b1)
        cnt += (gs[b1] != 0.f && ms[b1] >= t) ? 1 : 0;
    red[tid] = (float)BB * L1v[(size_t)tid * DD + d] +
               dfv[(size_t)tid * DD + d] * (float)cnt;
    __syncthreads();
    for (int s = 64; s > 0; s >>= 1) {
        if (tid < s) red[tid] += red[tid + s];
        __syncthreads();
    }
    if (tid == 0) partial[d] = red[0];
}

// ---------------- final scalar ----------------
__global__ __launch_bounds__(256) void final_k(const float* __restrict__ partial,
                                               float* __restrict__ out) {
    __shared__ float red[DD];
    const int tid = threadIdx.x;
    red[tid] = partial[tid];
    __syncthreads();
    for (int s = 128; s > 0; s >>= 1) {
        if (tid < s) red[tid] += red[tid + s];
        __syncthreads();
    }
    if (tid == 0) out[0] = red[0] * (1.0f / ((float)BB * (float)BB * (float)DD));
}

extern "C" void kernel_launch(void* const* d_in, const int* in_sizes, int n_in,
                              void* d_out, int out_size, void* d_ws, size_t ws_size,
                              hipStream_t stream) {
    const float* weight = (const float*)d_in[0];            // [C,D]
    const float* mu     = (const float*)d_in[1];            // [B,D]
    const float* var    = (const float*)d_in[2];            // [B,D]
    const int*   labels = (const int*)d_in[3];              // [B]
    const unsigned char* nontrivial = (const unsigned char*)d_in[4]; // [B,K,D] bool
    float* out = (float*)d_out;

    // scratch carve-up (floats)
    float* ws      = (float*)d_ws;
    float* sw      = ws;                          // B*D
    float* wnorm   = sw + BB * DD;                // C
    float* swnorm  = wnorm + CC;                  // B
    float* dis     = swnorm + BB;                 // B*C
    int*   idxbuf  = (int*)(dis + (size_t)BB * CC); // B*K
    float* L1v     = (float*)(idxbuf + BB * KK);  // B*D
    float* dfv     = L1v + BB * DD;               // B*D
    float* mval    = dfv + BB * DD;               // B*D
    float* totv    = mval + BB * DD;              // B*D
    float* gval    = totv + BB * DD;              // B*D
    float* partial = gval + BB * DD;              // D

    prep_norms <<<CC / 256, 256, 0, stream>>>(weight, wnorm);
    prep_sw    <<<BB, DD, 0, stream>>>(weight, labels, wnorm, sw, swnorm);
    gemm_dis   <<<(BB / 16) * (CC / 16), 32, 0, stream>>>(weight, sw, wnorm, swnorm, dis);
    sort_topk  <<<BB, 512, 0, stream>>>(dis, idxbuf);
    stats_kernel<<<BB, DD, 0, stream>>>(weight, sw, mu, var, labels, nontrivial,
                                        idxbuf, L1v, dfv, mval, totv, gval);
    reduce_d   <<<DD, BB, 0, stream>>>(L1v, dfv, mval, totv, gval, partial);
    final_k    <<<1, DD, 0, stream>>>(partial, out);
}